// Block_49830210568598
// MI455X (gfx1250) — compile-verified
//
#include <hip/hip_runtime.h>
#include <cmath>

typedef __attribute__((ext_vector_type(2))) float v2f;
typedef __attribute__((ext_vector_type(8))) float v8f;

#define HWT 9216   // 96*96 pixels per image
#define IMG 96

__device__ __forceinline__ float gelu_exact(float x) {
    return 0.5f * x * (1.0f + erff(x * 0.70710678118654752440f));
}

// ---------------------------------------------------------------------------
// LayerNorm over channel dim, NCHW layout. One thread per (b, hw) pixel.
// in is pre-offset for channel base; in_bstride is batch stride of input.
// out is contiguous (B, C, HWT).
// ---------------------------------------------------------------------------
__global__ void ln_kernel(const float* __restrict__ in, long in_bstride,
                          const float* __restrict__ g, const float* __restrict__ bta,
                          float* __restrict__ out, int C, int total) {
    int idx = blockIdx.x * blockDim.x + threadIdx.x;
    if (idx >= total) return;
    int b  = idx / HWT;
    int hw = idx - b * HWT;
    const float* p = in + (long)b * in_bstride + hw;
    float s1 = 0.f, s2 = 0.f;
    for (int c = 0; c < C; ++c) { float v = p[(long)c * HWT]; s1 += v; s2 += v * v; }
    float mu  = s1 / (float)C;
    float var = s2 / (float)C - mu * mu;
    float rs  = rsqrtf(var + 1e-5f);
    float* o = out + (long)b * C * HWT + hw;
    for (int c = 0; c < C; ++c) {
        float v = p[(long)c * HWT];
        o[(long)c * HWT] = (v - mu) * rs * g[c] + bta[c];
    }
}

// ---------------------------------------------------------------------------
// Generic conv1x1 as WMMA fp32 GEMM:
//   out[b, oc, hw] = act( sum_ic W[oc,ic] * X[b,ic,hw] + bias[oc] ) + resid[b,oc,hw]
// One wave computes one 16x16 (oc, hw) tile with V_WMMA_F32_16X16X4_F32.
// grid = (HWT/16, OC/16, B), block = 32.
// act: 0 none, 1 relu, 2 gelu
// ---------------------------------------------------------------------------
__global__ __launch_bounds__(32)
void gemm1x1_kernel(const float* __restrict__ W, const float* __restrict__ X,
                    const float* __restrict__ bias, const float* __restrict__ resid,
                    float* __restrict__ out,
                    int OC, int IC, long xbs, long rbs, long obs, int act) {
    int lane = threadIdx.x;
    int hl   = lane >> 4;       // half-wave: selects K pair {0,1} vs {2,3}
    int l16  = lane & 15;
    int hw0  = blockIdx.x * 16;
    int oc0  = blockIdx.y * 16;
    int b    = blockIdx.z;

    const float* Xc = X + (long)b * xbs + hw0 + l16;       // one column per lane
    const float* Wr = W + (long)(oc0 + l16) * IC;          // one row per lane

    v8f c = {0.f, 0.f, 0.f, 0.f, 0.f, 0.f, 0.f, 0.f};
    for (int k = 0; k < IC; k += 4) {
        int ka = k + hl * 2;
        v2f a, bb;
        a.x  = Wr[ka];
        a.y  = Wr[ka + 1];
        bb.x = Xc[(long)ka * HWT];
        bb.y = Xc[(long)(ka + 1) * HWT];
        c = __builtin_amdgcn_wmma_f32_16x16x4_f32(false, a, false, bb,
                                                  (short)0, c, false, false);
    }
#pragma unroll
    for (int v = 0; v < 8; ++v) {
        int oc = oc0 + v + hl * 8;
        int hw = hw0 + l16;
        float val = c[v];
        if (bias)      val += bias[oc];
        if (act == 1)      val = fmaxf(val, 0.f);
        else if (act == 2) val = gelu_exact(val);
        if (resid) val += resid[(long)b * rbs + (long)oc * HWT + hw];
        out[(long)b * obs + (long)oc * HWT + hw] = val;
    }
}

// ---------------------------------------------------------------------------
// Depthwise 3x3, zero pad, optional bias and GELU. One thread per element.
// ---------------------------------------------------------------------------
__global__ void dw3x3_kernel(const float* __restrict__ in, const float* __restrict__ w,
                             const float* __restrict__ bias, float* __restrict__ out,
                             int C, int n, int act) {
    int idx = blockIdx.x * blockDim.x + threadIdx.x;
    if (idx >= n) return;
    int hw = idx % HWT;
    int t  = idx / HWT;
    int c  = t % C;
    int b  = t / C;
    int y = hw / IMG, x = hw % IMG;
    const float* ip = in + ((long)b * C + c) * HWT;
    const float* wp = w + c * 9;
    float s = 0.f;
    for (int ti = 0; ti < 3; ++ti) {
        int yy = y + ti - 1;
        if (yy < 0 || yy >= IMG) continue;
        for (int tj = 0; tj < 3; ++tj) {
            int xx = x + tj - 1;
            if (xx < 0 || xx >= IMG) continue;
            s += ip[yy * IMG + xx] * wp[ti * 3 + tj];
        }
    }
    if (bias) s += bias[c];
    if (act)  s = gelu_exact(s);
    out[idx] = s;
}

// ---------------------------------------------------------------------------
// Full 3x3 conv, 16->16 channels (ada_p2). One thread per output element.
// ---------------------------------------------------------------------------
__global__ void conv3x3_16_kernel(const float* __restrict__ in, const float* __restrict__ w,
                                  const float* __restrict__ bias, float* __restrict__ out,
                                  int n) {
    int idx = blockIdx.x * blockDim.x + threadIdx.x;
    if (idx >= n) return;
    int hw = idx % HWT;
    int t  = idx / HWT;
    int oc = t % 16;
    int b  = t / 16;
    int y = hw / IMG, x = hw % IMG;
    const float* ip = in + (long)b * 16 * HWT;
    float s = bias[oc];
    for (int ic = 0; ic < 16; ++ic) {
        const float* wp = w + (oc * 16 + ic) * 9;
        for (int ti = 0; ti < 3; ++ti) {
            int yy = y + ti - 1;
            if (yy < 0 || yy >= IMG) continue;
            for (int tj = 0; tj < 3; ++tj) {
                int xx = x + tj - 1;
                if (xx < 0 || xx >= IMG) continue;
                s += ip[(long)ic * HWT + yy * IMG + xx] * wp[ti * 3 + tj];
            }
        }
    }
    out[idx] = s;
}

// ---------------------------------------------------------------------------
// Elementwise a *= b
// ---------------------------------------------------------------------------
__global__ void mul_kernel(float* __restrict__ a, const float* __restrict__ b, int n) {
    int idx = blockIdx.x * blockDim.x + threadIdx.x;
    if (idx < n) a[idx] *= b[idx];
}

// ---------------------------------------------------------------------------
// Adaptive conv:
//   out[b, g*12+o, p] = gelu( sum_r para[b,r,p] * sum_d ada_w[g,o,d,r]*xu[b,g,d,p] )
// One wave per (p-tile of 16, g, b). xu (108x16) and para (16x16) staged in LDS;
// 12 o-blocks, each a K=108 fp32 WMMA GEMM t[r,p], then r-contraction with para
// via per-lane dot + cross-half shuffle reduce.
// grid = (576, 16, 2), block = 32.
// ---------------------------------------------------------------------------
__global__ __launch_bounds__(32)
void ada_kernel(const float* __restrict__ m2, const float* __restrict__ para,
                const float* __restrict__ adaw, float* __restrict__ out) {
    __shared__ float xu[108 * 16];
    __shared__ float pa[16 * 16];
    int lane = threadIdx.x;
    int p0 = blockIdx.x * 16;
    int g  = blockIdx.y;
    int b  = blockIdx.z;

    // Build unfolded xu tile: d = cl*9 + (i*3+j), neighborhood of channel g*12+cl
    for (int idx = lane; idx < 108 * 16; idx += 32) {
        int d = idx >> 4, col = idx & 15;
        int cl = d / 9, nn = d % 9;
        int di = nn / 3 - 1, dj = nn % 3 - 1;
        int p = p0 + col;
        int y = p / IMG + di, x = p % IMG + dj;
        float v = 0.f;
        if (y >= 0 && y < IMG && x >= 0 && x < IMG)
            v = m2[((long)b * 192 + g * 12 + cl) * HWT + y * IMG + x];
        xu[idx] = v;
    }
    for (int idx = lane; idx < 256; idx += 32) {
        int r = idx >> 4, col = idx & 15;
        pa[idx] = para[((long)b * 16 + r) * HWT + p0 + col];
    }
    __syncthreads();

    int hl = lane >> 4, l16 = lane & 15;
    for (int o = 0; o < 12; ++o) {
        v8f c = {0.f, 0.f, 0.f, 0.f, 0.f, 0.f, 0.f, 0.f};
        const float* wa = adaw + (long)(g * 12 + o) * 108 * 16;  // [d][r]
        for (int k = 0; k < 108; k += 4) {
            int ka = k + hl * 2;
            v2f a, bb;
            a.x  = wa[ka * 16 + l16];          // A[r=l16][d=ka] = ada_w[g,o,ka,r]
            a.y  = wa[(ka + 1) * 16 + l16];
            bb.x = xu[ka * 16 + l16];          // B[d][p]
            bb.y = xu[(ka + 1) * 16 + l16];
            c = __builtin_amdgcn_wmma_f32_16x16x4_f32(false, a, false, bb,
                                                      (short)0, c, false, false);
        }
        float part = 0.f;
#pragma unroll
        for (int v = 0; v < 8; ++v) part += c[v] * pa[(v + hl * 8) * 16 + l16];
        part += __shfl_xor(part, 16, 32);
        if (hl == 0)
            out[((long)b * 192 + g * 12 + o) * HWT + p0 + l16] = gelu_exact(part);
    }
}

// ---------------------------------------------------------------------------
// Shifted-window attention. One block of 128 threads (4 waves) per
// (window, batch, head). The +/-SHIFT rolls are folded into gather/scatter:
// source/destination pixel = ((wy*8+i+4)%96, (wx*8+j+4)%96).
// sim = WMMA q.k^T (K=32) + rel-pos bias + shift mask -> softmax -> WMMA attn.v.
// grid = (144, 2, 3), block = 128.
// ---------------------------------------------------------------------------
__global__ __launch_bounds__(128)
void attn_kernel(const float* __restrict__ qkv, const float* __restrict__ relpos,
                 float* __restrict__ out) {
    __shared__ float qb[64 * 32];
    __shared__ float kb[64 * 32];
    __shared__ float vb[64 * 32];
    __shared__ float sb[64 * 64];
    int tid = threadIdx.x;
    int w = blockIdx.x, b = blockIdx.y, h = blockIdx.z;
    int wy = w / 12, wx = w % 12;

    for (int idx = tid; idx < 3 * 2048; idx += 128) {
        int which = idx / 2048;
        int rem = idx & 2047;
        int p = rem >> 5, d = rem & 31;
        int i = p >> 3, j = p & 7;
        int ys = (wy * 8 + i + 4) % IMG;
        int xs = (wx * 8 + j + 4) % IMG;
        float v = qkv[((long)b * 288 + which * 96 + h * 32 + d) * HWT + ys * IMG + xs];
        float* dst = (which == 0) ? qb : (which == 1) ? kb : vb;
        dst[rem] = v;
    }
    __syncthreads();

    int wv = tid >> 5;
    int lane = tid & 31;
    int hl = lane >> 4, l16 = lane & 15;
    int r0 = wv * 16;
    const float scale = 0.17677669529663687f;  // 32^-0.5

    // sim rows r0..r0+15 over 4 column tiles
    for (int ct = 0; ct < 4; ++ct) {
        v8f c = {0.f, 0.f, 0.f, 0.f, 0.f, 0.f, 0.f, 0.f};
        for (int k = 0; k < 32; k += 4) {
            int ka = k + hl * 2;
            v2f a, bb;
            a.x  = qb[(r0 + l16) * 32 + ka];
            a.y  = qb[(r0 + l16) * 32 + ka + 1];
            bb.x = kb[(ct * 16 + l16) * 32 + ka];   // B[k][n] = K[n][k]
            bb.y = kb[(ct * 16 + l16) * 32 + ka + 1];
            c = __builtin_amdgcn_wmma_f32_16x16x4_f32(false, a, false, bb,
                                                      (short)0, c, false, false);
        }
#pragma unroll
        for (int v = 0; v < 8; ++v) {
            int p = r0 + v + hl * 8;
            int q = ct * 16 + l16;
            int ip = p >> 3, jp = p & 7, iq = q >> 3, jq = q & 7;
            int rel = (ip - iq + 7) * 15 + (jp - jq + 7);
            float val = c[v] * scale + relpos[rel * 3 + h];
            bool msk = false;
            if (wy == 11 && ((ip < 4) != (iq < 4))) msk = true;
            if (wx == 11 && ((jp < 4) != (jq < 4))) msk = true;
            sb[p * 64 + q] = msk ? -INFINITY : val;
        }
    }
    __syncthreads();

    // softmax per row (lanes 0..15 of each wave handle its 16 rows)
    if (lane < 16) {
        int r = r0 + lane;
        float m = -INFINITY;
        for (int q = 0; q < 64; ++q) m = fmaxf(m, sb[r * 64 + q]);
        float s = 0.f;
        for (int q = 0; q < 64; ++q) s += expf(sb[r * 64 + q] - m);
        float inv = 1.f / s;
        for (int q = 0; q < 64; ++q) sb[r * 64 + q] = expf(sb[r * 64 + q] - m) * inv;
    }
    __syncthreads();

    // out rows = attn (16x64) x v (64x32), scattered un-rolled
    for (int ct = 0; ct < 2; ++ct) {
        v8f c = {0.f, 0.f, 0.f, 0.f, 0.f, 0.f, 0.f, 0.f};
        for (int k = 0; k < 64; k += 4) {
            int ka = k + hl * 2;
            v2f a, bb;
            a.x  = sb[(r0 + l16) * 64 + ka];
            a.y  = sb[(r0 + l16) * 64 + ka + 1];
            bb.x = vb[ka * 32 + ct * 16 + l16];
            bb.y = vb[(ka + 1) * 32 + ct * 16 + l16];
            c = __builtin_amdgcn_wmma_f32_16x16x4_f32(false, a, false, bb,
                                                      (short)0, c, false, false);
        }
#pragma unroll
        for (int v = 0; v < 8; ++v) {
            int p = r0 + v + hl * 8;
            int d = ct * 16 + l16;
            int i = p >> 3, j = p & 7;
            int yo = (wy * 8 + i + 4) % IMG;
            int xo = (wx * 8 + j + 4) % IMG;
            out[((long)b * 96 + h * 32 + d) * HWT + yo * IMG + xo] = c[v];
        }
    }
}

// ---------------------------------------------------------------------------
extern "C" void kernel_launch(void* const* d_in, const int* in_sizes, int n_in,
                              void* d_out, int out_size, void* d_ws, size_t ws_size,
                              hipStream_t stream) {
    (void)in_sizes; (void)n_in; (void)out_size; (void)ws_size;
    const float* x        = (const float*)d_in[0];
    const float* cb_ln_g  = (const float*)d_in[1];
    const float* cb_ln_b  = (const float*)d_in[2];
    const float* cb_c1_pw = (const float*)d_in[3];
    const float* cb_c1_dw = (const float*)d_in[4];
    const float* cb_c2_pw = (const float*)d_in[5];
    const float* ada_w    = (const float*)d_in[6];
    const float* ada_p1_w = (const float*)d_in[7];
    const float* ada_p1_b = (const float*)d_in[8];
    const float* ada_p2_w = (const float*)d_in[9];
    const float* ada_p2_b = (const float*)d_in[10];
    const float* cb_c3_w  = (const float*)d_in[11];
    const float* ln1_g    = (const float*)d_in[12];
    const float* ln1_b    = (const float*)d_in[13];
    const float* qkv_w    = (const float*)d_in[14];
    const float* qkv_b    = (const float*)d_in[15];
    const float* rel_pos  = (const float*)d_in[16];
    const float* proj_w   = (const float*)d_in[17];
    const float* proj_b   = (const float*)d_in[18];
    const float* ln2_g    = (const float*)d_in[19];
    const float* ln2_b    = (const float*)d_in[20];
    const float* ffn_w1   = (const float*)d_in[21];
    const float* ffn_b1   = (const float*)d_in[22];
    const float* ffn_dw   = (const float*)d_in[23];
    const float* ffn_db   = (const float*)d_in[24];
    const float* ffn_w2   = (const float*)d_in[25];
    const float* ffn_b2   = (const float*)d_in[26];
    const float* out_w    = (const float*)d_in[27];
    const float* out_b    = (const float*)d_in[28];
    float* outp = (float*)d_out;

    const long HW = HWT;
    float* ws  = (float*)d_ws;
    float* R   = ws;                 // (B,96,HW)   1,769,472  LN scratch (reused 3x)
    float* M1  = R   + 1769472;      // (B,192,HW)  3,538,944  m1 / ada_out / attn_out
    float* B1  = M1  + 3538944;      // (B,192,HW)  3,538,944  dw(m1) / prod / f1dw
    float* M2  = B1  + 3538944;      // (B,192,HW)  3,538,944  m2 / ffn f1
    float* P1  = M2  + 3538944;      // (B,16,HW)     294,912
    float* P2  = P1  + 294912;       // (B,16,HW)     294,912
    float* CAT = P2  + 294912;       // (B,192,HW)  3,538,944  concat buffer
    float* QKV = CAT + 3538944;      // (B,288,HW)  5,308,416
    float* XT1 = QKV + 5308416;      // (B,96,HW)   1,769,472

    dim3 b32(32), b128(128), b256(256);
    int npix = 2 * HWT;              // 18432
    int nMID = 2 * 192 * HWT;        // 3,538,944

    // ===== Conv branch =====
    ln_kernel<<<dim3(npix / 256), b256, 0, stream>>>(x, 192 * HW, cb_ln_g, cb_ln_b, R, 96, npix);
    gemm1x1_kernel<<<dim3(576, 12, 2), b32, 0, stream>>>(cb_c1_pw, R, nullptr, nullptr, M1,
                                                         192, 96, 96 * HW, 0, 192 * HW, 0);
    gemm1x1_kernel<<<dim3(576, 12, 2), b32, 0, stream>>>(cb_c2_pw, R, nullptr, nullptr, M2,
                                                         192, 96, 96 * HW, 0, 192 * HW, 0);
    dw3x3_kernel<<<dim3(nMID / 256), b256, 0, stream>>>(M1, cb_c1_dw, nullptr, B1, 192, nMID, 0);
    gemm1x1_kernel<<<dim3(576, 1, 2), b32, 0, stream>>>(ada_p1_w, M2, ada_p1_b, nullptr, P1,
                                                        16, 192, 192 * HW, 0, 16 * HW, 1);
    conv3x3_16_kernel<<<dim3(2 * 16 * HWT / 256), b256, 0, stream>>>(P1, ada_p2_w, ada_p2_b, P2,
                                                                     2 * 16 * HWT);
    ada_kernel<<<dim3(576, 16, 2), b32, 0, stream>>>(M2, P2, ada_w, M1);   // ada_out (gelu'd)
    mul_kernel<<<dim3(nMID / 256), b256, 0, stream>>>(B1, M1, nMID);       // prod = b1*b2
    gemm1x1_kernel<<<dim3(576, 6, 2), b32, 0, stream>>>(cb_c3_w, B1, nullptr, x, CAT,
                                                        96, 192, 192 * HW, 192 * HW, 192 * HW, 0);

    // ===== Transformer branch =====
    ln_kernel<<<dim3(npix / 256), b256, 0, stream>>>(x + 96 * HW, 192 * HW, ln1_g, ln1_b, R, 96, npix);
    gemm1x1_kernel<<<dim3(576, 18, 2), b32, 0, stream>>>(qkv_w, R, qkv_b, nullptr, QKV,
                                                         288, 96, 96 * HW, 0, 288 * HW, 0);
    attn_kernel<<<dim3(144, 2, 3), b128, 0, stream>>>(QKV, rel_pos, M1);
    gemm1x1_kernel<<<dim3(576, 6, 2), b32, 0, stream>>>(proj_w, M1, proj_b, x + 96 * HW, XT1,
                                                        96, 96, 96 * HW, 192 * HW, 96 * HW, 0);
    ln_kernel<<<dim3(npix / 256), b256, 0, stream>>>(XT1, 96 * HW, ln2_g, ln2_b, R, 96, npix);
    gemm1x1_kernel<<<dim3(576, 12, 2), b32, 0, stream>>>(ffn_w1, R, ffn_b1, nullptr, M2,
                                                         192, 96, 96 * HW, 0, 192 * HW, 0);
    dw3x3_kernel<<<dim3(nMID / 256), b256, 0, stream>>>(M2, ffn_dw, ffn_db, B1, 192, nMID, 1);
    gemm1x1_kernel<<<dim3(576, 6, 2), b32, 0, stream>>>(ffn_w2, B1, ffn_b2, XT1, CAT + 96 * HW,
                                                        96, 192, 192 * HW, 96 * HW, 192 * HW, 0);

    // ===== Fuse: out = x + out_w . concat + out_b =====
    gemm1x1_kernel<<<dim3(576, 12, 2), b32, 0, stream>>>(out_w, CAT, out_b, x, outp,
                                                         192, 192, 192 * HW, 192 * HW, 192 * HW, 0);
}